// HistogramLoss_26886495272980
// MI455X (gfx1250) — compile-verified
//
#include <hip/hip_runtime.h>
#include <hip/hip_bf16.h>

#define BINS 64
#define NROWS 96                       // B*C = 32*3
#define ROW_ELEMS (512 * 512)          // 262144 elements per (b,c) row
#define BLOCKS_PER_ROW 32
#define THREADS 256
#define NWAVES (THREADS / 32)
#define VEC4_PER_BLOCK (ROW_ELEMS / 4 / BLOCKS_PER_ROW)  // 2048 float4 per block

typedef __attribute__((ext_vector_type(2))) float v2f;
typedef __attribute__((ext_vector_type(8))) float v8f;

__global__ void zero_counts_kernel(unsigned int* __restrict__ p, int n) {
    int i = blockIdx.x * blockDim.x + threadIdx.x;
    if (i < n) p[i] = 0u;
}

__device__ __forceinline__ int bin_of(float x) {
    // torch.histc semantics on [0,1]: idx = clip(int(x*64), 0, 63)
    x = fminf(fmaxf(x, 0.0f), 1.0f);
    int idx = (int)(x * 64.0f);
    return idx > 63 ? 63 : idx;
}

// Per-wave privatized LDS histograms (ds_add_u32), b128 global loads.
__global__ void __launch_bounds__(THREADS)
hist_kernel(const float* __restrict__ x, unsigned int* __restrict__ counts) {
    __shared__ unsigned int lhist[NWAVES * BINS];
    const int tid  = threadIdx.x;
    const int wave = tid >> 5;

    for (int i = tid; i < NWAVES * BINS; i += THREADS) lhist[i] = 0u;
    __syncthreads();

    const int row   = blockIdx.x / BLOCKS_PER_ROW;
    const int chunk = blockIdx.x % BLOCKS_PER_ROW;
    const float4* __restrict__ p =
        reinterpret_cast<const float4*>(x + (size_t)row * ROW_ELEMS) +
        (size_t)chunk * VEC4_PER_BLOCK;

    unsigned int* wh = &lhist[wave * BINS];
    for (int i = tid; i < VEC4_PER_BLOCK; i += THREADS) {
        float4 v = p[i];
        atomicAdd(&wh[bin_of(v.x)], 1u);
        atomicAdd(&wh[bin_of(v.y)], 1u);
        atomicAdd(&wh[bin_of(v.z)], 1u);
        atomicAdd(&wh[bin_of(v.w)], 1u);
    }
    __syncthreads();

    if (tid < BINS) {
        unsigned int s = 0;
#pragma unroll
        for (int w = 0; w < NWAVES; ++w) s += lhist[w * BINS + tid];
        atomicAdd(&counts[row * BINS + tid], s);
    }
}

// One wave. 64 partial L1 sums (2 per lane) reduced with a single
// V_WMMA_F32_16X16X4_F32 against an all-ones B: D[i][j] = sum_k A[i][k],
// so per-lane sum of the 8 D regs + lane^16 partner = grand total.
__global__ void __launch_bounds__(32)
finalize_kernel(const unsigned int* __restrict__ cf,
                const unsigned int* __restrict__ cr,
                float* __restrict__ out) {
    const int   lane  = threadIdx.x;         // exactly 32 threads, EXEC all ones
    const int   N     = NROWS * BINS;        // 6144
    const float invHW = 1.0f / (float)ROW_ELEMS;

    float a0 = 0.0f, a1 = 0.0f;
    for (int i = lane; i < N; i += 64)
        a0 += fabsf((float)(int)cf[i] - (float)(int)cr[i]);
    for (int i = lane + 32; i < N; i += 64)
        a1 += fabsf((float)(int)cf[i] - (float)(int)cr[i]);

    v2f A; A.x = a0 * invHW; A.y = a1 * invHW;   // 64 distinct A elements across wave
    v2f B; B.x = 1.0f;       B.y = 1.0f;         // all-ones 4x16 B
    v8f C = {};
    // (neg_a, A, neg_b, B, c_mod, C, reuse_a, reuse_b)
    v8f D = __builtin_amdgcn_wmma_f32_16x16x4_f32(
        false, A, false, B, (short)0, C, false, false);

    float s = D[0] + D[1] + D[2] + D[3] + D[4] + D[5] + D[6] + D[7];
    s += __shfl_xor(s, 16, 32);                  // rows 0-7 + rows 8-15
    if (lane == 0) out[0] = s / (float)N;        // mean over B*C*BINS
}

extern "C" void kernel_launch(void* const* d_in, const int* in_sizes, int n_in,
                              void* d_out, int out_size, void* d_ws, size_t ws_size,
                              hipStream_t stream) {
    const float* fake = (const float*)d_in[0];
    const float* real = (const float*)d_in[1];

    unsigned int* cf = (unsigned int*)d_ws;       // 96*64 u32
    unsigned int* cr = cf + NROWS * BINS;         // 96*64 u32

    const int total_counts = 2 * NROWS * BINS;    // 12288
    zero_counts_kernel<<<(total_counts + THREADS - 1) / THREADS, THREADS, 0, stream>>>(
        cf, total_counts);

    dim3 grid(NROWS * BLOCKS_PER_ROW);            // 3072 blocks per input
    hist_kernel<<<grid, THREADS, 0, stream>>>(fake, cf);
    hist_kernel<<<grid, THREADS, 0, stream>>>(real, cr);

    finalize_kernel<<<1, 32, 0, stream>>>(cf, cr, (float*)d_out);
}